// RegionLoss_69277822485107
// MI455X (gfx1250) — compile-verified
//
#include <hip/hip_runtime.h>
#include <hip/hip_bf16.h>
#include <math.h>

// ---------------- problem constants (from the reference) ----------------
constexpr int NB = 64, NA = 5, NC = 80, NH = 38, NW = 38, MAXGT = 50;
constexpr int CH    = 5 + NC;          // 85 channels per anchor
constexpr int PLANE = NH * NW;         // 1444
constexpr int TOTCELLS = NB * NA * PLANE;   // 462080
#define LN80F 4.3820266346738812f

typedef __attribute__((ext_vector_type(2))) float v2f;
typedef __attribute__((ext_vector_type(8))) float v8f;

__device__ __forceinline__ float sigm(float x) { return 1.0f / (1.0f + expf(-x)); }

// bbox_iou_center, literally the reference formula (including its unit mixing)
__device__ __forceinline__ float iou_center(float x1, float y1, float w1, float h1,
                                            float x2, float y2, float w2, float h2) {
    float uw = fmaxf(x1 + w1 * 0.5f, x2 + w2 * 0.5f) - fminf(x1 - w1 * 0.5f, x2 - w2 * 0.5f);
    float uh = fmaxf(y1 + h1 * 0.5f, y2 + h2 * 0.5f) - fminf(y1 - h1 * 0.5f, y2 - h2 * 0.5f);
    float cw = w1 + w2 - uw;
    float ch = h1 + h2 - uh;
    float inter = (cw > 0.0f && ch > 0.0f) ? cw * ch : 0.0f;
    return inter / (w1 * h1 + w2 * h2 - inter);
}

// ---------------------------------------------------------------------------
// Kernel A: per-batch GT processing + masked-cell loss terms (<=3200 cells).
// Adds to ws[0]: for each owned masked cell,
//   (x-tx)^2+(y-ty)^2+(w-tw)^2+(h-th)^2
//   + 25*(conf-tconf)^2 - conf^2          (correction vs. kernel B's conf^2)
//   + (lse - picked) - ln(80)             (correction vs. kernel C's base)
// Adds to ws[1]: nGT.
// ---------------------------------------------------------------------------
__global__ void __launch_bounds__(64)
gt_masked_kernel(const float* __restrict__ out, const float* __restrict__ tgt,
                 const float* __restrict__ anchors, float* __restrict__ ws) {
    __shared__ float st[MAXGT * 5];
    __shared__ int   scell[MAXGT];
    __shared__ int   sV;

    const int b   = blockIdx.x;
    const int tid = threadIdx.x;

    for (int i = tid; i < MAXGT * 5; i += blockDim.x) st[i] = tgt[b * MAXGT * 5 + i];
    __syncthreads();

    if (tid == 0) {                    // valid = cumprod(gxn != 0): leading run
        int V = MAXGT;
        for (int t = 0; t < MAXGT; ++t)
            if (st[t * 5 + 1] == 0.0f) { V = t; break; }
        sV = V;
    }
    __syncthreads();
    const int V = sV;

    int   bn = 0, gi = 0, gj = 0;
    float gw = 0.f, gh = 0.f, gxn = 0.f, gyn = 0.f, clsf = 0.f;

    if (tid < MAXGT) {
        if (tid < V) {
            clsf = st[tid * 5 + 0];
            gxn  = st[tid * 5 + 1];
            gyn  = st[tid * 5 + 2];
            gw   = st[tid * 5 + 3] * (float)NW;
            gh   = st[tid * 5 + 4] * (float)NH;
            gi = (int)(gxn * (float)NW);
            gj = (int)(gyn * (float)NH);
            float best = -1e30f;
            #pragma unroll
            for (int a = 0; a < NA; ++a) {   // first-max wins, like jnp.argmax
                float aw = anchors[2 * a], ah = anchors[2 * a + 1];
                float v  = iou_center(0.f, 0.f, gw, gh, 0.f, 0.f, aw, ah);
                if (v > best) { best = v; bn = a; }
            }
            scell[tid] = bn * PLANE + gj * NW + gi;
        } else {
            scell[tid] = -1000 - tid;    // never matches a valid cell
        }
    }
    __syncthreads();

    if (tid < V) {
        // scatter-duplicate resolution: last valid GT with this cell owns it
        bool owner = true;
        for (int t2 = tid + 1; t2 < V; ++t2)
            if (scell[t2] == scell[tid]) { owner = false; break; }

        if (owner) {
            const float aw = anchors[2 * bn], ah = anchors[2 * bn + 1];
            const int base = ((b * NA + bn) * CH) * PLANE + gj * NW + gi;
            const float o0 = out[base + 0 * PLANE];
            const float o1 = out[base + 1 * PLANE];
            const float o2 = out[base + 2 * PLANE];
            const float o3 = out[base + 3 * PLANE];
            const float o4 = out[base + 4 * PLANE];

            // pred box (reference: normalized units) vs gt_full (cell units)
            const float px = ((float)gi + sigm(o0)) / (float)NW;
            const float py = ((float)gj + sigm(o1)) / (float)NH;
            const float pw = expf(o2) * aw / (float)NW;
            const float ph = expf(o3) * ah / (float)NH;
            const float tconf = iou_center(gxn * (float)NW, gyn * (float)NH, gw, gh,
                                           px, py, pw, ph);

            const float tx = gxn * (float)NW - (float)gi;
            const float ty = gyn * (float)NH - (float)gj;
            const float tw = logf(gw / aw);
            const float th = logf(gh / ah);
            const int   tc = (int)clsf;

            const float x = sigm(o0), y = sigm(o1);
            float contrib = (x - tx) * (x - tx) + (y - ty) * (y - ty)
                          + (o2 - tw) * (o2 - tw) + (o3 - th) * (o3 - th);

            const float conf  = sigm(o4);
            const float dconf = conf - tconf;
            contrib += 25.0f * dconf * dconf - conf * conf;   // minus kernel-B term

            // class logsumexp over the 80 class channels (mask==1 -> raw logits)
            float m = -INFINITY, picked = 0.0f;
            for (int c = 0; c < NC; ++c) {
                float l = out[base + (5 + c) * PLANE];
                m = fmaxf(m, l);
                if (c == tc) picked = l;
            }
            float s = 0.0f;
            for (int c = 0; c < NC; ++c)
                s += expf(out[base + (5 + c) * PLANE] - m);
            contrib += (m + logf(s)) - picked - LN80F;        // minus kernel-C base

            atomicAdd(&ws[0], contrib);
        }
    }
    if (tid == 0) atomicAdd(&ws[1], (float)V);
}

// ---------------------------------------------------------------------------
// Kernel B: sum sigmoid(conf)^2 over all cells. One block per (b,a) plane,
// coalesced reads of the contiguous 1444-float conf plane.
// Per-wave reduction uses V_WMMA_F32_16X16X4_F32 as an exact fp32 accumulator:
//   A = ones(16x4), B vgpr0 = per-lane value, vgpr1 = 0
//   => D[m][n] += value[lane n] + value[lane n+16]  (accumulated in C)
// After the loop, lanes 0..15 of acc[0] hold the 16 column partials; a 4-step
// xor-shuffle finishes the wave, then one atomicAdd per wave.
// ---------------------------------------------------------------------------
__global__ void __launch_bounds__(256)
conf_stream_kernel(const float* __restrict__ out, float* __restrict__ ws) {
    const int ba  = blockIdx.x;           // 0..NB*NA-1
    const int tid = threadIdx.x;
    const float* plane = out + ((size_t)ba * CH + 4) * PLANE;

    v8f acc = {};
    const v2f aones = {1.0f, 1.0f};

    #pragma unroll
    for (int it = 0; it < 6; ++it) {      // 6*256 = 1536 >= 1444
        const int p = tid + it * 256;
        float v = 0.0f;
        if (p < PLANE) v = plane[p];
        float s  = sigm(v);
        float sq = (p < PLANE) ? s * s : 0.0f;   // inactive lanes contribute 0
        v2f bop = {sq, 0.0f};
        // wave-uniform control flow here: EXEC is all-ones as WMMA requires
        acc = __builtin_amdgcn_wmma_f32_16x16x4_f32(
                  /*neg_a=*/false, aones, /*neg_b=*/false, bop,
                  /*c_mod=*/(short)0, acc, /*reuse_a=*/false, /*reuse_b=*/false);
    }

    float w = acc[0];                      // D[0][n] (lanes 16..31 duplicate it)
    #pragma unroll
    for (int off = 8; off >= 1; off >>= 1)
        w += __shfl_xor(w, off, 32);       // reduce the 16 columns
    if ((tid & 31) == 0) atomicAdd(&ws[0], w);
}

// ---------------------------------------------------------------------------
// Kernel C: finalize. Base term = 462080 * ln(80) (every unmasked cell's exact
// class contribution when logits are all zero), then divide by nGT.
// ---------------------------------------------------------------------------
__global__ void finalize_kernel(const float* __restrict__ ws, float* __restrict__ outp) {
    outp[0] = (ws[0] + (float)TOTCELLS * LN80F) / ws[1];
}

extern "C" void kernel_launch(void* const* d_in, const int* in_sizes, int n_in,
                              void* d_out, int out_size, void* d_ws, size_t ws_size,
                              hipStream_t stream) {
    const float* out_t   = (const float*)d_in[0];   // (64, 425, 38, 38) fp32
    const float* target  = (const float*)d_in[1];   // (64, 250) fp32
    const float* anchors = (const float*)d_in[2];   // (5, 2) fp32
    float* ws = (float*)d_ws;                       // ws[0]=loss sum, ws[1]=nGT

    hipMemsetAsync(d_ws, 0, 2 * sizeof(float), stream);
    gt_masked_kernel<<<NB, 64, 0, stream>>>(out_t, target, anchors, ws);
    conf_stream_kernel<<<NB * NA, 256, 0, stream>>>(out_t, ws);
    finalize_kernel<<<1, 1, 0, stream>>>(ws, (float*)d_out);
}